// MyGNN_45397804319006
// MI455X (gfx1250) — compile-verified
//
#include <hip/hip_runtime.h>

typedef float v2f __attribute__((ext_vector_type(2)));
typedef float v8f __attribute__((ext_vector_type(8)));

#define D 128
#define LDSS 132   // padded LDS row stride (floats): bank-conflict-free fragment reads

// ---------------- zero scratch ----------------
__global__ __launch_bounds__(256) void zero_f32(float* __restrict__ p, int n4) {
  int i = blockIdx.x * 256 + threadIdx.x;
  if (i < n4) ((float4*)p)[i] = make_float4(0.f, 0.f, 0.f, 0.f);
}

// ---------------- edge scatter-sum (GCN aggregate) ----------------
// one wave32 per edge; each lane moves 4 consecutive floats -> coalesced
// gather + hardware global_atomic_add_f32 (unsafeAtomicAdd).
__global__ __launch_bounds__(256) void scatter_add_f32(
    const float* __restrict__ feat, const int* __restrict__ src,
    const int* __restrict__ dst, float* __restrict__ agg, int E) {
  int t = blockIdx.x * 256 + threadIdx.x;
  int e = t >> 5;
  if (e >= E) return;
  int lane = t & 31;
  int s = src[e];
  int d = dst[e];
  const float4 v = *(const float4*)(feat + (size_t)s * D + lane * 4);
  float* p = agg + (size_t)d * D + lane * 4;
  unsafeAtomicAdd(p + 0, v.x);
  unsafeAtomicAdd(p + 1, v.y);
  unsafeAtomicAdd(p + 2, v.z);
  unsafeAtomicAdd(p + 3, v.w);
}

// ---------------- WMMA f32 16x16x4 tile GEMM core ----------------
// Block = 256 threads = 8 waves. LDS holds one 16x128 A tile (padded).
// Wave w computes the 16x16 output tile at columns [16w, 16w+16).
// Exact fp32 MAC via V_WMMA_F32_16X16X4_F32, K-loop of 32 steps.
__device__ __forceinline__ void tile_gemm_store(
    const float* As, const float* __restrict__ W,
    const float* __restrict__ bias, float* __restrict__ outBase) {
  const int tid  = threadIdx.x;
  const int lane = tid & 31;
  const int w    = tid >> 5;
  const int cb   = w * 16;
  const int r    = lane & 15;         // A row (M) for this lane
  const int hi   = lane >> 4;         // lane-half selects K pair / M+8
  const int koff = hi << 1;
  const int c    = lane & 15;         // B/D column (N) for this lane
  v8f acc = {};
#pragma unroll
  for (int kb = 0; kb < D; kb += 4) {
    v2f a = *(const v2f*)(As + r * LDSS + kb + koff);   // A[r][kb+koff+0/1]
    v2f b;
    b.x = W[(kb + koff) * D + cb + c];                   // B[k][n] row-major
    b.y = W[(kb + koff + 1) * D + cb + c];
    acc = __builtin_amdgcn_wmma_f32_16x16x4_f32(
        /*neg_a=*/false, a, /*neg_b=*/false, b,
        /*c_mod=*/(short)0, acc, /*reuse_a=*/false, /*reuse_b=*/false);
  }
#pragma unroll
  for (int v = 0; v < 8; ++v) {
    int row = v + (hi << 3);          // C/D layout: M = v + 8*hi, N = c
    int col = cb + c;
    outBase[(size_t)row * D + col] = acc[v] + bias[col];
  }
}

// C[rowbase:rowbase+16, :] = A[rowbase:rowbase+16, :] @ W + bias
__global__ __launch_bounds__(256) void gemm_bias(
    const float* __restrict__ A, const float* __restrict__ W,
    const float* __restrict__ bias, float* __restrict__ C) {
  __shared__ float As[16 * LDSS];
  const int rowbase = blockIdx.x * 16;
  const int tid = threadIdx.x;
  const int e0  = tid * 8;            // 2048 floats / 256 threads
  const int row = e0 >> 7;
  const int col = e0 & 127;
  const float* src = A + (size_t)(rowbase + row) * D + col;
  *(float4*)(As + row * LDSS + col)     = *(const float4*)(src);
  *(float4*)(As + row * LDSS + col + 4) = *(const float4*)(src + 4);
  __syncthreads();
  tile_gemm_store(As, W, bias, C + (size_t)rowbase * D);
}

// out[rowbase:rowbase+16, :] = feat[bidx*adj+nidx][:] @ W + bias  (2D gather)
__global__ __launch_bounds__(256) void gather_gemm_bias(
    const float* __restrict__ feat, const int* __restrict__ bidx,
    const int* __restrict__ nidx, const float* __restrict__ W,
    const float* __restrict__ bias, float* __restrict__ out, int adj) {
  __shared__ float As[16 * LDSS];
  const int rowbase = blockIdx.x * 16;
  const int tid = threadIdx.x;
  const int e0  = tid * 8;
  const int row = e0 >> 7;
  const int col = e0 & 127;
  const int rsrc = bidx[rowbase + row] * adj + nidx[rowbase + row];
  const float* src = feat + (size_t)rsrc * D + col;
  *(float4*)(As + row * LDSS + col)     = *(const float4*)(src);
  *(float4*)(As + row * LDSS + col + 4) = *(const float4*)(src + 4);
  __syncthreads();
  tile_gemm_store(As, W, bias, out + (size_t)rowbase * D);
}

extern "C" void kernel_launch(void* const* d_in, const int* in_sizes, int n_in,
                              void* d_out, int out_size, void* d_ws, size_t ws_size,
                              hipStream_t stream) {
  const float* x0    = (const float*)d_in[0];
  const float* x1    = (const float*)d_in[1];
  const int*   edge  = (const int*)d_in[2];   // [2,E]: src = edge, dst = edge+E
  const int*   b_i0  = (const int*)d_in[3];
  const int*   n_i0  = (const int*)d_in[4];
  const int*   b_i1  = (const int*)d_in[5];
  const int*   n_i1  = (const int*)d_in[6];
  const float* W1    = (const float*)d_in[7];
  const float* bias1 = (const float*)d_in[8];
  const float* W2    = (const float*)d_in[9];
  const float* bias2 = (const float*)d_in[10];
  const float* Wl    = (const float*)d_in[11];
  const float* bl    = (const float*)d_in[12];
  const float* Wfi   = (const float*)d_in[13];
  const float* bfi   = (const float*)d_in[14];
  float* out = (float*)d_out;

  const int Nn  = in_sizes[0] / D;   // 64000 nodes
  const int E   = in_sizes[2] / 2;   // 1M edges
  const int M   = in_sizes[3];       // 32000 gathered rows
  const int ADJ = 1000;

  float* agg = (float*)d_ws;               // [Nn*D] f32  (32 MB)
  float* h   = agg + (size_t)Nn * D;       // [Nn*D] f32  (32 MB)

  const int zero4  = Nn * D / 4;
  const int zblk   = (zero4 + 255) / 256;
  const int sblk   = (int)(((size_t)E * 32 + 255) / 256);
  dim3 blk(256);

  // ---- GCN layer 1: agg = segsum(x1[src] -> dst); h = agg@W1 + b1
  zero_f32<<<zblk, blk, 0, stream>>>(agg, zero4);
  scatter_add_f32<<<sblk, blk, 0, stream>>>(x1, edge, edge + E, agg, E);
  gemm_bias<<<Nn / 16, blk, 0, stream>>>(agg, W1, bias1, h);

  // ---- GCN layer 2: agg = segsum(h[src] -> dst); h = agg@W2 + b2
  zero_f32<<<zblk, blk, 0, stream>>>(agg, zero4);
  scatter_add_f32<<<sblk, blk, 0, stream>>>(h, edge, edge + E, agg, E);
  gemm_bias<<<Nn / 16, blk, 0, stream>>>(agg, W2, bias2, h);

  // ---- branch 0: d0 = x0[b0,n0] @ Wfi + bfi
  gather_gemm_bias<<<M / 16, blk, 0, stream>>>(x0, b_i0, n_i0, Wfi, bfi,
                                               out, ADJ);
  // ---- branch 1: d1 = h[b1,n1] @ Wl + bl
  gather_gemm_bias<<<M / 16, blk, 0, stream>>>(h, b_i1, n_i1, Wl, bl,
                                               out + (size_t)M * D, ADJ);
}